// SIGNConvolution_3135326126433
// MI455X (gfx1250) — compile-verified
//
#include <hip/hip_runtime.h>

typedef __attribute__((ext_vector_type(2))) float v2f;
typedef __attribute__((ext_vector_type(4))) float v4f;
typedef __attribute__((ext_vector_type(8))) float v8f;
typedef __attribute__((ext_vector_type(4))) unsigned int u32x4;
typedef __attribute__((ext_vector_type(8))) int i32x8;
typedef __attribute__((ext_vector_type(4))) int i32x4;

#define N_NODES_C 100000
#define N_IN_C    256
#define N_OUT_C   64
#define N_EDGES_C 1600000
#define W_ELEMS   (N_IN_C * N_OUT_C)   // 16384 floats = 64 KB

// ---------------------------------------------------------------------------
// Zero d_out (harness poisons it with 0xAA before timing).
// ---------------------------------------------------------------------------
__global__ __launch_bounds__(256) void sign_zero_kernel(v4f* __restrict__ out, int n4) {
  int i = blockIdx.x * blockDim.x + threadIdx.x;
  if (i < n4) {
    v4f z = {0.f, 0.f, 0.f, 0.f};
    out[i] = z;
  }
}

// ---------------------------------------------------------------------------
// h = x @ W + b via V_WMMA_F32_16X16X4_F32.
//  - Wave 0 DMAs W (64 KB) into LDS with the Tensor Data Mover
//    (tensor_load_to_lds + s_wait_tensorcnt), then the block re-lays it out
//    k-pair-interleaved + XOR-16 swizzled so every B fragment is ONE
//    bank-conflict-free ds_load_b64 (no re-pairing v_movs before WMMA).
//  - Block = 128 threads = 4 waves; each wave owns a 16x64 h tile
//    (4 x v8f accumulators), bias fused into accumulator init.
//  - 100000 % 16 == 0 -> no ragged tiles, EXEC stays all-ones for WMMA.
// ---------------------------------------------------------------------------
__global__ __launch_bounds__(128) void sign_gemm_kernel(const float* __restrict__ x,
                                                        const float* __restrict__ W,
                                                        const float* __restrict__ bias,
                                                        float* __restrict__ h) {
  __shared__ float lds_raw[W_ELEMS];  // TDM destination: W row-major
  __shared__ float lds_b[W_ELEMS];    // interleaved: [(k/2)*128 + ((n^swz)*2) + (k&1)]

  const int tid = threadIdx.x;

  // ---- TDM: DMA W -> lds_raw (one wave issues; EXEC ignored by TDM) ----
  if (tid < 32) {
    const unsigned long long gaddr = (unsigned long long)(uintptr_t)W;
    const unsigned int lds_off = (unsigned int)(uintptr_t)(void*)lds_raw; // LDS byte offset
    u32x4 g0;
    g0[0] = 1u;                                  // count=1, user descriptor
    g0[1] = lds_off;                             // lds_addr
    g0[2] = (unsigned int)gaddr;                 // global_addr[31:0]
    g0[3] = (unsigned int)((gaddr >> 32) & 0x01FFFFFFu) | 0x80000000u; // addr[56:32] | type=2
    i32x8 g1;
    g1[0] = 0x00020000;                          // workgroup_mask=0, data_size=4B
    g1[1] = (int)0x40000000;                     // tensor_dim0 = 16384 (bits[31:16])
    g1[2] = 0x00010000;                          // tensor_dim1 = 1
    g1[3] = (int)0x40000000;                     // tile_dim0 = 16384 (bits[31:16])
    g1[4] = 1;                                   // tile_dim1 = 1, tile_dim2 = 0
    g1[5] = 16384;                               // tensor_dim0_stride
    g1[6] = 0;
    g1[7] = 0;
    i32x4 gz4 = {0, 0, 0, 0};                    // groups 2/3 unused (2D tensor)
    i32x8 gz8 = {0, 0, 0, 0, 0, 0, 0, 0};        // extra group (clang-23 6-arg form)
    __builtin_amdgcn_tensor_load_to_lds(g0, g1, gz4, gz4, gz8, 0);
    __builtin_amdgcn_s_wait_tensorcnt(0);
  }
  __syncthreads();

  // ---- interleave into B-fragment layout: one ds_load_b64 per fragment ----
  for (int i = tid * 4; i < W_ELEMS; i += 128 * 4) {
    const int k   = i >> 6;                      // W row
    const int n   = i & 63;                      // W col (multiple of 4)
    const int swz = ((k >> 1) & 1) << 4;         // bank swizzle per k-pair
    v4f w = *(const v4f*)&lds_raw[i];
    float* dst = &lds_b[(k >> 1) * 128 + ((n ^ swz) * 2) + (k & 1)];
    dst[0] = w.x; dst[2] = w.y; dst[4] = w.z; dst[6] = w.w;
  }
  __syncthreads();

  const int wave = tid >> 5;
  const int lane = tid & 31;
  const int tile = blockIdx.x * 4 + wave;        // 16-row tile index
  if (tile >= N_NODES_C / 16) return;            // uniform per-wave guard

  const int row0 = tile * 16;
  const int m    = lane & 15;                    // M (A) / N (B,C) within tile
  const int koff = (lane >> 4) * 2;              // K sub-offset: 0 or 2

  v8f acc[4];
#pragma unroll
  for (int t = 0; t < 4; ++t) {
    float bv = bias[t * 16 + m];
#pragma unroll
    for (int i = 0; i < 8; ++i) acc[t][i] = bv;
  }

  const float* xrow = x + (size_t)(row0 + m) * N_IN_C + koff;

#pragma unroll 8
  for (int k = 0; k < N_IN_C; k += 4) {
    v2f a = *(const v2f*)(xrow + k);             // global_load_b64
    const int pr  = (k + koff) >> 1;             // k-pair row (0..127)
    const int swz = (pr & 1) << 4;
    const float* bp = &lds_b[pr * 128];
#pragma unroll
    for (int t = 0; t < 4; ++t) {
      v2f bm = *(const v2f*)&bp[((t * 16 + m) ^ swz) * 2];   // ds_load_b64
      acc[t] = __builtin_amdgcn_wmma_f32_16x16x4_f32(
          /*neg_a=*/false, a, /*neg_b=*/false, bm,
          /*c_mod=*/(short)0, acc[t], /*reuse_a=*/false, /*reuse_b=*/false);
    }
  }

  // C/D layout: VGPR i -> M = i + 8*(lane>=16), N = (lane&15) + t*16.
  const int mstore = (lane >> 4) * 8;
#pragma unroll
  for (int i = 0; i < 8; ++i) {
    const size_t rbase = (size_t)(row0 + mstore + i) * N_OUT_C;
#pragma unroll
    for (int t = 0; t < 4; ++t) {
      h[rbase + t * 16 + m] = acc[t][i];
    }
  }
}

// ---------------------------------------------------------------------------
// out[row[e]] += adj[e] * h[col[e]]   (h and out both resident in 192MB L2)
// One wave per 4 edges; wave id forced uniform via readfirstlane so edge
// metadata becomes scalar (SMEM) loads, keeping the vector path free for the
// coalesced 256B h-gather + hardware global_atomic_add_f32 (2 cols/lane).
// ---------------------------------------------------------------------------
__global__ __launch_bounds__(256) void sign_scatter_kernel(const float* __restrict__ h,
                                                           const float* __restrict__ adj,
                                                           const int*   __restrict__ erow,
                                                           const int*   __restrict__ ecol,
                                                           float* __restrict__ out) {
  const unsigned gtid = blockIdx.x * blockDim.x + threadIdx.x;
  const unsigned wv   = (unsigned)__builtin_amdgcn_readfirstlane((int)(gtid >> 5));
  const unsigned e0   = wv * 4;
  if (e0 >= N_EDGES_C) return;
  const int co = (threadIdx.x & 31) * 2;

#pragma unroll
  for (int j = 0; j < 4; ++j) {
    const unsigned e = e0 + j;
    const int   r = erow[e];                     // uniform -> s_load
    const int   c = ecol[e];                     // uniform -> s_load
    const float a = adj[e];                      // uniform -> s_load
    v2f v = *(const v2f*)&h[(size_t)c * N_OUT_C + co];   // global_load_b64
    float* o = &out[(size_t)r * N_OUT_C + co];
    unsafeAtomicAdd(o,     v.x * a);             // global_atomic_add_f32
    unsafeAtomicAdd(o + 1, v.y * a);
  }
}

// ---------------------------------------------------------------------------
// Launch: zero(out) -> gemm(h in d_ws) -> scatter(out). Stream-ordered.
// ---------------------------------------------------------------------------
extern "C" void kernel_launch(void* const* d_in, const int* in_sizes, int n_in,
                              void* d_out, int out_size, void* d_ws, size_t ws_size,
                              hipStream_t stream) {
  const float* x    = (const float*)d_in[0];
  const float* W    = (const float*)d_in[1];
  const float* bias = (const float*)d_in[2];
  const float* adj  = (const float*)d_in[3];
  const int*   erow = (const int*)d_in[4];
  const int*   ecol = (const int*)d_in[5];
  float* out = (float*)d_out;
  float* h   = (float*)d_ws;   // 100000*64*4 = 25.6 MB scratch

  // 1) zero out (6.4M floats -> 1.6M v4f stores)
  const int n4 = (N_NODES_C * N_OUT_C) / 4;
  sign_zero_kernel<<<(n4 + 255) / 256, 256, 0, stream>>>((v4f*)out, n4);

  // 2) h = x@W + b : 6250 row-tiles, 4 per block
  const int ntiles = N_NODES_C / 16;             // 6250
  sign_gemm_kernel<<<(ntiles + 3) / 4, 128, 0, stream>>>(x, W, bias, h);

  // 3) edge scatter: one wave per 4 edges (400k waves, 12.8M threads)
  const long long nthreads = (long long)(N_EDGES_C / 4) * 32;
  sign_scatter_kernel<<<(int)(nthreads / 256), 256, 0, stream>>>(h, adj, erow, ecol, out);
}